// Seq2Seq_34857954575082
// MI455X (gfx1250) — compile-verified
//
#include <hip/hip_runtime.h>
#include <math.h>

// Problem constants (match reference)
#define B_ 128
#define S_ 50
#define T_ 50
#define V_ 10000
#define E_ 128
#define H_ 128
#define NEG_ (-1.0e10f)

typedef float v2f __attribute__((ext_vector_type(2)));
typedef float v8f __attribute__((ext_vector_type(8)));

// ---------------------------------------------------------------------------
// CDNA5 native fp32 WMMA: D(16x16) = A(16x4) * B(4x16) + C
// ---------------------------------------------------------------------------
__device__ __forceinline__ v8f wmma4(v2f a, v2f b, v8f c) {
  return __builtin_amdgcn_wmma_f32_16x16x4_f32(
      /*neg_a=*/false, a, /*neg_b=*/false, b,
      /*c_mod=*/(short)0, c, /*reuse_a=*/false, /*reuse_b=*/false);
}

// One 16x16 f32 tile: sum_k A[m][k] * W[n*ldw + k]   (i.e. A @ W^T tile)
// A staged in LDS with row stride lda (pad by 4 -> conflict-free ds_load_b64).
// K is split into two halves feeding TWO independent accumulator chains to
// halve the WMMA D->C dependency depth; summed exactly at the end.
// 'ncol' is this lane's absolute output column n = nbase + (lane&15).
__device__ __forceinline__ v8f wmma_tile_accum(const float* lA, int lda,
                                               const float* W, int ldw,
                                               int K, int ncol) {
  const int lane = threadIdx.x & 31;
  const int hi   = lane >> 4;
  const float* ap = lA + (lane & 15) * lda + 2 * hi;          // k = kk + 2*hi + {0,1}
  const float* bp = W + (size_t)ncol * ldw + 2 * hi;
  const int Kh = K >> 1;                                       // K % 8 == 0 everywhere
  v8f acc0 = {0.f, 0.f, 0.f, 0.f, 0.f, 0.f, 0.f, 0.f};
  v8f acc1 = {0.f, 0.f, 0.f, 0.f, 0.f, 0.f, 0.f, 0.f};
  for (int kk = 0; kk < Kh; kk += 4) {
    v2f a0 = *(const v2f*)(ap + kk);
    v2f b0 = *(const v2f*)(bp + kk);
    v2f a1 = *(const v2f*)(ap + Kh + kk);
    v2f b1 = *(const v2f*)(bp + Kh + kk);
    acc0 = wmma4(a0, b0, acc0);
    acc1 = wmma4(a1, b1, acc1);
  }
  return acc0 + acc1;
}

// Store a 16x16 C tile (layout: lane l, vgpr v -> m = v + 8*(l>>4), n = l&15)
// act: 0 = none, 1 = tanh
__device__ __forceinline__ void store_tile(float* out, size_t ldo,
                                           int mbase, int nbase, v8f c,
                                           const float* bias, int act) {
  const int lane = threadIdx.x & 31;
  const int hi   = lane >> 4;
  const int n    = nbase + (lane & 15);
  const float bv = bias ? bias[n] : 0.f;
#pragma unroll
  for (int v = 0; v < 8; ++v) {
    float x = c[v] + bv;
    if (act == 1) x = tanhf(x);
    out[(size_t)(mbase + v + 8 * hi) * ldo + n] = x;
  }
}

__device__ __forceinline__ float sigm(float x) { return 1.f / (1.f + expf(-x)); }

// ---------------------------------------------------------------------------
// Init kernels
// ---------------------------------------------------------------------------
__global__ void zero_kernel(float* p, size_t n) {
  size_t i = (size_t)blockIdx.x * blockDim.x + threadIdx.x;
  if (i < n) p[i] = 0.f;
}

__global__ void zero_out_t0_kernel(float* out) {
  int i = blockIdx.x * 256 + threadIdx.x;
  if (i < B_ * V_) {
    int b = i / V_, n = i % V_;
    out[(size_t)b * T_ * V_ + n] = 0.f;
  }
}

// ---------------------------------------------------------------------------
// K1: gather embeddings + input-gate GEMM for BOTH encoder directions.
// gi_dir[(s*B+b)*3H + n] = emb(src[b,s]) @ Wih_dir^T + bih_dir
// grid (6, 400, 2), block 128 (4 waves, 1 n-tile each)
// ---------------------------------------------------------------------------
__global__ void embed_gi_kernel(const int* __restrict__ src,
                                const float* __restrict__ enc_emb,
                                const float* __restrict__ Wih_f,
                                const float* __restrict__ bih_f,
                                const float* __restrict__ Wih_b,
                                const float* __restrict__ bih_b,
                                float* __restrict__ gi_f,
                                float* __restrict__ gi_b) {
  __shared__ __align__(16) float eA[16 * (E_ + 4)];
  const int tid   = threadIdx.x;
  const int mbase = blockIdx.y * 16;          // m = s*B + b
  const int dir   = blockIdx.z;
  const int s     = mbase >> 7;               // / B_
#pragma unroll
  for (int i = 0; i < 16; ++i) {
    const int b   = (mbase + i) & (B_ - 1);
    const int tok = src[b * S_ + s];
    eA[i * (E_ + 4) + tid] = enc_emb[(size_t)tok * E_ + tid];
  }
  __syncthreads();
  const float* W    = dir ? Wih_b : Wih_f;
  const float* bias = dir ? bih_b : bih_f;
  float*       gi   = dir ? gi_b : gi_f;
  const int wave  = tid >> 5;
  const int nbase = (blockIdx.x * 4 + wave) * 16;   // 24 n-tiles
  v8f acc = wmma_tile_accum(eA, E_ + 4, W, E_, E_, nbase + (tid & 15));
  store_tile(gi, 3 * H_, mbase, nbase, acc, bias, 0);
}

// ---------------------------------------------------------------------------
// K2: one recurrent encoder step, both directions (grid (8, 2), block 128).
// gh = h @ Whh^T + bhh via WMMA into LDS, then elementwise GRU update.
// ---------------------------------------------------------------------------
__global__ void enc_step_kernel(const float* __restrict__ gi_f,
                                const float* __restrict__ gi_b,
                                const float* __restrict__ Whh_f,
                                const float* __restrict__ bhh_f,
                                const float* __restrict__ Whh_b,
                                const float* __restrict__ bhh_b,
                                float* __restrict__ h_f,
                                float* __restrict__ h_b,
                                float* __restrict__ enc_out, int t) {
  __shared__ __align__(16) float hA[16 * (H_ + 4)];
  __shared__ __align__(16) float ghL[16 * 388];
  const int tid = threadIdx.x;
  const int dir = blockIdx.y;
  const int s   = dir ? (S_ - 1 - t) : t;
  const float* gi  = dir ? gi_b : gi_f;
  const float* Whh = dir ? Whh_b : Whh_f;
  const float* bhh = dir ? bhh_b : bhh_f;
  float* h = dir ? h_b : h_f;
  const int mbase = blockIdx.x * 16;
#pragma unroll
  for (int i = 0; i < 16; ++i)
    hA[i * (H_ + 4) + tid] = h[(size_t)(mbase + i) * H_ + tid];
  __syncthreads();
  const int wave = tid >> 5;
#pragma unroll
  for (int j = 0; j < 6; ++j) {                       // 24 n-tiles / 4 waves
    const int nbase = (wave * 6 + j) * 16;
    v8f acc = wmma_tile_accum(hA, H_ + 4, Whh, H_, H_, nbase + (tid & 15));
    store_tile(ghL, 388, 0, nbase, acc, bhh, 0);
  }
  __syncthreads();
  const int c = tid;
#pragma unroll 4
  for (int r = 0; r < 16; ++r) {
    const int b = mbase + r;
    const size_t g = ((size_t)s * B_ + b) * (3 * H_);
    float ir = gi[g + c], iz = gi[g + H_ + c], in_ = gi[g + 2 * H_ + c];
    float hr = ghL[r * 388 + c], hz = ghL[r * 388 + H_ + c], hn = ghL[r * 388 + 2 * H_ + c];
    float rg = sigm(ir + hr);
    float zg = sigm(iz + hz);
    float ng = tanhf(in_ + rg * hn);
    float hp = hA[r * (H_ + 4) + c];
    float h2 = (1.f - zg) * ng + zg * hp;
    h[(size_t)b * H_ + c] = h2;
    enc_out[((size_t)b * S_ + s) * (2 * H_) + dir * H_ + c] = h2;
  }
}

// ---------------------------------------------------------------------------
// K3: hidden = tanh([h_f,h_b] @ Wfc^T + bfc).  grid (2, 8)
// ---------------------------------------------------------------------------
__global__ void fc_hidden_kernel(const float* __restrict__ h_f,
                                 const float* __restrict__ h_b,
                                 const float* __restrict__ Wfc,
                                 const float* __restrict__ bfc,
                                 float* __restrict__ h_dec) {
  __shared__ __align__(16) float aA[16 * (2 * H_ + 4)];
  const int tid = threadIdx.x;
  const int mbase = blockIdx.y * 16;
#pragma unroll
  for (int i = 0; i < 32; ++i) {
    const int lin = tid + i * 128;
    const int r = lin >> 8, c = lin & 255;
    aA[r * (2 * H_ + 4) + c] = (c < H_)
        ? h_f[(size_t)(mbase + r) * H_ + c]
        : h_b[(size_t)(mbase + r) * H_ + (c - H_)];
  }
  __syncthreads();
  const int wave  = tid >> 5;
  const int nbase = (blockIdx.x * 4 + wave) * 16;
  v8f acc = wmma_tile_accum(aA, 2 * H_ + 4, Wfc, 2 * H_, 2 * H_, nbase + (tid & 15));
  store_tile(h_dec, H_, mbase, nbase, acc, bfc, 1);
}

// ---------------------------------------------------------------------------
// K4: step-invariant attention term (HOISTED out of decoder loop):
// enc_part[(b*S+s)*H + n] = enc_out[b,s,:] @ attn_W[:,H:3H]^T + attn_b
// grid (2, 400)
// ---------------------------------------------------------------------------
__global__ void enc_part_kernel(const float* __restrict__ enc_out,
                                const float* __restrict__ attn_W,
                                const float* __restrict__ attn_b,
                                float* __restrict__ enc_part) {
  __shared__ __align__(16) float aA[16 * (2 * H_ + 4)];
  const int tid = threadIdx.x;
  const int mbase = blockIdx.y * 16;                 // m = b*S + s
#pragma unroll
  for (int i = 0; i < 32; ++i) {
    const int lin = tid + i * 128;
    const int r = lin >> 8, c = lin & 255;
    aA[r * (2 * H_ + 4) + c] = enc_out[(size_t)(mbase + r) * (2 * H_) + c];
  }
  __syncthreads();
  const int wave  = tid >> 5;
  const int nbase = (blockIdx.x * 4 + wave) * 16;
  v8f acc = wmma_tile_accum(aA, 2 * H_ + 4, attn_W + H_, 3 * H_, 2 * H_,
                            nbase + (tid & 15));
  store_tile(enc_part, H_, mbase, nbase, acc, attn_b, 0);
}

// ---------------------------------------------------------------------------
// K5: h_part = h_dec @ attn_W[:,0:H]^T.  grid (2, 8)
// ---------------------------------------------------------------------------
__global__ void hpart_kernel(const float* __restrict__ h_dec,
                             const float* __restrict__ attn_W,
                             float* __restrict__ h_part) {
  __shared__ __align__(16) float aA[16 * (H_ + 4)];
  const int tid = threadIdx.x;
  const int mbase = blockIdx.y * 16;
#pragma unroll
  for (int i = 0; i < 16; ++i)
    aA[i * (H_ + 4) + tid] = h_dec[(size_t)(mbase + i) * H_ + tid];
  __syncthreads();
  const int wave  = tid >> 5;
  const int nbase = (blockIdx.x * 4 + wave) * 16;
  v8f acc = wmma_tile_accum(aA, H_ + 4, attn_W, 3 * H_, H_, nbase + (tid & 15));
  store_tile(h_part, H_, mbase, nbase, acc, nullptr, 0);
}

// ---------------------------------------------------------------------------
// K6: decoder embedding gather + attention scores + masked softmax + context.
// One block per batch row, 128 threads (4 waves, wave32 shfl reductions).
// Writes pred_in[b] = [ h2(later) | embd | context ]  (512 floats)
// ---------------------------------------------------------------------------
__global__ void dec_attn_kernel(const int* __restrict__ trg,
                                const int* __restrict__ src_len,
                                const float* __restrict__ dec_emb,
                                const float* __restrict__ h_part,
                                const float* __restrict__ enc_part,
                                const float* __restrict__ attn_v,
                                const float* __restrict__ enc_out,
                                float* __restrict__ pred_in, int t) {
  __shared__ float sc[S_];
  __shared__ float awS[S_];
  const int b = blockIdx.x;
  const int tid = threadIdx.x;
  const int lane = tid & 31;
  const int wave = tid >> 5;
  {
    const int tok = trg[b * T_ + t];
    pred_in[(size_t)b * 512 + H_ + tid] = dec_emb[(size_t)tok * E_ + tid];
  }
  const float hv0 = h_part[(size_t)b * H_ + lane];
  const float hv1 = h_part[(size_t)b * H_ + lane + 32];
  const float hv2 = h_part[(size_t)b * H_ + lane + 64];
  const float hv3 = h_part[(size_t)b * H_ + lane + 96];
  const float av0 = attn_v[lane],      av1 = attn_v[lane + 32];
  const float av2 = attn_v[lane + 64], av3 = attn_v[lane + 96];
  for (int s = wave; s < S_; s += 4) {
    const float* ep = enc_part + ((size_t)b * S_ + s) * H_;
    float v = av0 * tanhf(hv0 + ep[lane])
            + av1 * tanhf(hv1 + ep[lane + 32])
            + av2 * tanhf(hv2 + ep[lane + 64])
            + av3 * tanhf(hv3 + ep[lane + 96]);
#pragma unroll
    for (int off = 16; off >= 1; off >>= 1) v += __shfl_xor(v, off, 32);
    if (lane == 0) sc[s] = v;
  }
  __syncthreads();
  if (wave == 0) {                                   // masked softmax over S_=50
    const int slen = src_len[b];
    const int s1 = lane + 32;
    float v0 = (lane < slen) ? sc[lane] : NEG_;
    float v1 = (s1 < S_) ? ((s1 < slen) ? sc[s1] : NEG_) : NEG_;
    float mx = fmaxf(v0, v1);
#pragma unroll
    for (int off = 16; off >= 1; off >>= 1) mx = fmaxf(mx, __shfl_xor(mx, off, 32));
    float e0 = expf(v0 - mx);
    float e1 = (s1 < S_) ? expf(v1 - mx) : 0.f;
    float sm = e0 + e1;
#pragma unroll
    for (int off = 16; off >= 1; off >>= 1) sm += __shfl_xor(sm, off, 32);
    awS[lane] = e0 / sm;
    if (s1 < S_) awS[s1] = e1 / sm;
  }
  __syncthreads();
  float c0 = 0.f, c1 = 0.f;
  for (int s = 0; s < S_; ++s) {
    const float w = awS[s];
    const float* eo = enc_out + ((size_t)b * S_ + s) * (2 * H_);
    c0 += w * eo[tid];
    c1 += w * eo[H_ + tid];
  }
  pred_in[(size_t)b * 512 + 2 * H_ + tid] = c0;
  pred_in[(size_t)b * 512 + 3 * H_ + tid] = c1;
}

// ---------------------------------------------------------------------------
// K7: decoder GRU gates.  z=0: giD = rnn_in @ dec_Wih^T + bih (K=384)
//                         z=1: ghD = h_dec  @ dec_Whh^T + bhh (K=128)
// grid (6, 8, 2)
// ---------------------------------------------------------------------------
__global__ void dec_gemm_kernel(const float* __restrict__ pred_in,
                                const float* __restrict__ h_dec,
                                const float* __restrict__ dec_Wih,
                                const float* __restrict__ dec_bih,
                                const float* __restrict__ dec_Whh,
                                const float* __restrict__ dec_bhh,
                                float* __restrict__ giD,
                                float* __restrict__ ghD) {
  __shared__ __align__(16) float aA[16 * 388];
  const int tid = threadIdx.x;
  const int mbase = blockIdx.y * 16;
  const int z = blockIdx.z;
  if (z == 0) {
#pragma unroll
    for (int i = 0; i < 48; ++i) {
      const int lin = tid + i * 128;
      const int r = lin / 384, c = lin % 384;
      aA[r * 388 + c] = pred_in[(size_t)(mbase + r) * 512 + H_ + c];  // rnn_in
    }
  } else {
#pragma unroll
    for (int i = 0; i < 16; ++i)
      aA[i * 388 + tid] = h_dec[(size_t)(mbase + i) * H_ + tid];
  }
  __syncthreads();
  const int wave  = tid >> 5;
  const int nbase = (blockIdx.x * 4 + wave) * 16;     // 24 n-tiles
  if (z == 0) {
    v8f acc = wmma_tile_accum(aA, 388, dec_Wih, 3 * H_, 3 * H_, nbase + (tid & 15));
    store_tile(giD, 3 * H_, mbase, nbase, acc, dec_bih, 0);
  } else {
    v8f acc = wmma_tile_accum(aA, 388, dec_Whh, H_, H_, nbase + (tid & 15));
    store_tile(ghD, 3 * H_, mbase, nbase, acc, dec_bhh, 0);
  }
}

// ---------------------------------------------------------------------------
// K8: elementwise decoder GRU update; writes h2 into h_dec and pred_in[0:H].
// ---------------------------------------------------------------------------
__global__ void dec_gru_elem_kernel(const float* __restrict__ giD,
                                    const float* __restrict__ ghD,
                                    float* __restrict__ h_dec,
                                    float* __restrict__ pred_in) {
  const int idx = blockIdx.x * 128 + threadIdx.x;
  const int b = idx >> 7, c = idx & 127;
  const size_t g = (size_t)b * (3 * H_);
  float ir = giD[g + c], iz = giD[g + H_ + c], in_ = giD[g + 2 * H_ + c];
  float hr = ghD[g + c], hz = ghD[g + H_ + c], hn = ghD[g + 2 * H_ + c];
  float rg = sigm(ir + hr);
  float zg = sigm(iz + hz);
  float ng = tanhf(in_ + rg * hn);
  float hp = h_dec[(size_t)b * H_ + c];
  float h2 = (1.f - zg) * ng + zg * hp;
  h_dec[(size_t)b * H_ + c] = h2;
  pred_in[(size_t)b * 512 + c] = h2;
}

// ---------------------------------------------------------------------------
// K9: vocab projection  out[:, t+1, :] = pred_in @ Wout^T + bout
// M=128 (8 tiles), N=10000 (625 tiles, exact), K=512.  grid (40, 8),
// block = 4 waves, each wave owns 4 n-tiles -> 4 independent WMMA chains,
// A frag (from LDS) reused 4x per k-step.
// ---------------------------------------------------------------------------
__global__ void out_gemm_kernel(const float* __restrict__ pred_in,
                                const float* __restrict__ Wout,
                                const float* __restrict__ bout,
                                float* __restrict__ out, int t) {
  __shared__ __align__(16) float aA[16 * 516];
  const int tid = threadIdx.x;
  const int mbase = blockIdx.y * 16;
#pragma unroll
  for (int i = 0; i < 64; ++i) {
    const int lin = tid + i * 128;
    const int r = lin >> 9, c = lin & 511;
    aA[r * 516 + c] = pred_in[(size_t)(mbase + r) * 512 + c];
  }
  __syncthreads();
  const int lane = tid & 31, hi = lane >> 4, wave = tid >> 5;
  const int nt0 = blockIdx.x * 16 + wave * 4;
  const int nvalid = min(4, (V_ / 16) - nt0);          // 625 tiles total
  if (nvalid <= 0) return;                             // uniform per wave
  const float* ap = aA + (lane & 15) * 516 + 2 * hi;
  v8f acc[4];
  const float* bp[4];
#pragma unroll
  for (int j = 0; j < 4; ++j) {
    acc[j] = (v8f){0.f, 0.f, 0.f, 0.f, 0.f, 0.f, 0.f, 0.f};
    int n = (nt0 + j) * 16 + (lane & 15);
    if (n > V_ - 1) n = V_ - 1;                        // clamp: stay in-bounds
    bp[j] = Wout + (size_t)n * 512 + 2 * hi;
  }
  for (int kk = 0; kk < 512; kk += 4) {
    v2f a = *(const v2f*)(ap + kk);
#pragma unroll
    for (int j = 0; j < 4; ++j) {
      v2f bfrag = *(const v2f*)(bp[j] + kk);
      acc[j] = wmma4(a, bfrag, acc[j]);
    }
  }
  const size_t row_stride = (size_t)T_ * V_;
  const size_t tcol = (size_t)(t + 1) * V_;
  for (int j = 0; j < nvalid; ++j) {
    const int n = (nt0 + j) * 16 + (lane & 15);
    const float bv = bout[n];
#pragma unroll
    for (int v = 0; v < 8; ++v) {
      const int m = mbase + v + 8 * hi;
      out[(size_t)m * row_stride + tcol + n] = acc[j][v] + bv;
    }
  }
}

// ---------------------------------------------------------------------------
// Host side
// ---------------------------------------------------------------------------
extern "C" void kernel_launch(void* const* d_in, const int* in_sizes, int n_in,
                              void* d_out, int out_size, void* d_ws, size_t ws_size,
                              hipStream_t stream) {
  (void)in_sizes; (void)n_in; (void)out_size; (void)ws_size;
  const int*   src     = (const int*)d_in[0];
  const int*   src_len = (const int*)d_in[1];
  const int*   trg     = (const int*)d_in[2];
  const float* enc_emb = (const float*)d_in[3];
  const float* Wih_f   = (const float*)d_in[4];
  const float* Whh_f   = (const float*)d_in[5];
  const float* bih_f   = (const float*)d_in[6];
  const float* bhh_f   = (const float*)d_in[7];
  const float* Wih_b   = (const float*)d_in[8];
  const float* Whh_b   = (const float*)d_in[9];
  const float* bih_b   = (const float*)d_in[10];
  const float* bhh_b   = (const float*)d_in[11];
  const float* Wfc     = (const float*)d_in[12];
  const float* bfc     = (const float*)d_in[13];
  const float* attn_W  = (const float*)d_in[14];
  const float* attn_b  = (const float*)d_in[15];
  const float* attn_v  = (const float*)d_in[16];
  const float* dec_emb = (const float*)d_in[17];
  const float* dWih    = (const float*)d_in[18];
  const float* dWhh    = (const float*)d_in[19];
  const float* dbih    = (const float*)d_in[20];
  const float* dbhh    = (const float*)d_in[21];
  const float* Wout    = (const float*)d_in[22];
  const float* bout    = (const float*)d_in[23];
  float* out = (float*)d_out;

  // Workspace layout (~30.4 MB of fp32)
  float* ws = (float*)d_ws;
  float* gi_f     = ws; ws += (size_t)S_ * B_ * 3 * H_;
  float* gi_b     = ws; ws += (size_t)S_ * B_ * 3 * H_;
  float* enc_out  = ws; ws += (size_t)B_ * S_ * 2 * H_;
  float* enc_part = ws; ws += (size_t)B_ * S_ * H_;
  float* h_f      = ws; ws += B_ * H_;
  float* h_b      = ws; ws += B_ * H_;   // contiguous with h_f (zeroed together)
  float* h_dec    = ws; ws += B_ * H_;
  float* h_part   = ws; ws += B_ * H_;
  float* pred_in  = ws; ws += B_ * 512;
  float* giD      = ws; ws += B_ * 3 * H_;
  float* ghD      = ws; ws += B_ * 3 * H_;

  // Init: h0 = 0 for both encoder directions; outputs[:,0,:] = 0
  zero_kernel<<<(2 * B_ * H_ + 255) / 256, 256, 0, stream>>>(h_f, (size_t)2 * B_ * H_);
  zero_out_t0_kernel<<<(B_ * V_ + 255) / 256, 256, 0, stream>>>(out);

  // Encoder input gates (hoisted out of recurrence), both directions
  embed_gi_kernel<<<dim3(6, 400, 2), 128, 0, stream>>>(
      src, enc_emb, Wih_f, bih_f, Wih_b, bih_b, gi_f, gi_b);

  // Recurrent encoder chain (fwd + bwd per launch)
  for (int t = 0; t < S_; ++t)
    enc_step_kernel<<<dim3(8, 2), 128, 0, stream>>>(
        gi_f, gi_b, Whh_f, bhh_f, Whh_b, bhh_b, h_f, h_b, enc_out, t);

  // Decoder initial hidden + step-invariant attention term
  fc_hidden_kernel<<<dim3(2, 8), 128, 0, stream>>>(h_f, h_b, Wfc, bfc, h_dec);
  enc_part_kernel<<<dim3(2, 400), 128, 0, stream>>>(enc_out, attn_W, attn_b, enc_part);

  // Decoder chain
  for (int t = 0; t < T_ - 1; ++t) {
    hpart_kernel<<<dim3(2, 8), 128, 0, stream>>>(h_dec, attn_W, h_part);
    dec_attn_kernel<<<B_, 128, 0, stream>>>(
        trg, src_len, dec_emb, h_part, enc_part, attn_v, enc_out, pred_in, t);
    dec_gemm_kernel<<<dim3(6, 8, 2), 128, 0, stream>>>(
        pred_in, h_dec, dWih, dbih, dWhh, dbhh, giD, ghD);
    dec_gru_elem_kernel<<<B_, 128, 0, stream>>>(giD, ghD, h_dec, pred_in);
    out_gemm_kernel<<<dim3(40, 8), 128, 0, stream>>>(pred_in, Wout, bout, out, t);
  }
}